// CustomMaxPool2D_48137993453716
// MI455X (gfx1250) — compile-verified
//
#include <hip/hip_runtime.h>
#include <stdint.h>

typedef __attribute__((ext_vector_type(4))) float v4f;

namespace {
constexpr unsigned kPlanes  = 32 * 64;    // N*C = 2048
constexpr unsigned kIW      = 224;
constexpr unsigned kIH      = 224;
constexpr unsigned kOH      = 112;
constexpr unsigned kQPR     = 28;         // output quads (float4) per row
constexpr unsigned kThreads = 256;        // 8 wave32s
constexpr unsigned kIter    = 4;
constexpr unsigned kPlane0  = kPlanes / kIter;                 // 512
constexpr unsigned kTotalTh = kPlane0 * kOH * kQPR;            // 1,605,632
constexpr unsigned kBlocks  = kTotalTh / kThreads;             // 6272
static_assert(kBlocks * kThreads == kTotalTh, "exact tiling, no tail");

constexpr unsigned kPlaneBytes   = kIH * kIW * 4u;             // 200704
constexpr unsigned kInStepBytes  = kPlane0 * kPlaneBytes;      // 102,760,448 (<2^32)
constexpr unsigned kRowBytes     = kIW * 4u;                   // 896
constexpr unsigned kOutQuadsPerPlane = kOH * kQPR;             // 3136
constexpr unsigned kOutStepFloats    = kPlane0 * kOutQuadsPerPlane * 4;
// whole input = kPlanes*kPlaneBytes = 411,041,792 B < 4 GB -> 32-bit offsets OK
static_assert((unsigned long long)kPlanes * kPlaneBytes < (1ull << 32), "32-bit offsets");
}  // namespace

// 2x2 stride-2 max pool, fp32 NCHW. Per thread-iteration: 4x async NT b128
// global->LDS copies (GVS form: SGPR base + 32-bit VGPR offset, double
// buffered, ASYNCcnt-pipelined), 4x ds_load_b128 back, v_max3-friendly
// reduce, one NT b128 store. Iterations step along the plane axis so the
// index decomposition is done once; per-iteration addressing is one 32-bit add.
__global__ __launch_bounds__(kThreads) void maxpool2x2_s2_async_kernel(
    const float* __restrict__ x, float* __restrict__ out) {
  __shared__ __align__(16) float sbuf[2][kThreads][16];  // 32 KB, 2 stages

  const unsigned tid  = threadIdx.x;
  // Low 32 bits of a __shared__ generic address == LDS byte offset (ISA 10.2);
  // that is what the async-load VDST operand wants.
  const unsigned lds0 = (unsigned)(uintptr_t)(&sbuf[0][tid][0]);

  // One-time decomposition: g -> (plane0, oh, ow4).
  const unsigned g      = blockIdx.x * kThreads + tid;
  const unsigned ow4    = g % kQPR;
  const unsigned r      = g / kQPR;
  const unsigned oh     = r % kOH;
  const unsigned plane0 = r / kOH;

  // 32-bit byte offset of input row 2*oh for this thread's quad.
  unsigned inOff = plane0 * kPlaneBytes + (oh * 2u) * kRowBytes + ow4 * 32u;
  const unsigned long long xbase = (unsigned long long)(uintptr_t)x;

  float* op = out + ((unsigned long long)plane0 * kOutQuadsPerPlane +
                     (unsigned long long)oh * kQPR + ow4) * 4u;

  // Issue 4 async NT b128 global->LDS copies into `stage` (GVS addressing:
  // mem = SADDR + VADDR32 + inst_offset; inst_offset also biases the LDS
  // address, matching the packed LDS layout).
  auto issue = [&](unsigned off0, unsigned stage) {
    const unsigned l0   = lds0 + stage * (unsigned)sizeof(sbuf[0]);
    const unsigned l1   = l0 + 32u;
    const unsigned off1 = off0 + kRowBytes;  // next input row
    asm volatile("global_load_async_to_lds_b128 %0, %1, %2 th:TH_LOAD_NT"
                 :: "v"(l0), "v"(off0), "s"(xbase) : "memory");
    asm volatile("global_load_async_to_lds_b128 %0, %1, %2 offset:16 th:TH_LOAD_NT"
                 :: "v"(l0), "v"(off0), "s"(xbase) : "memory");
    asm volatile("global_load_async_to_lds_b128 %0, %1, %2 th:TH_LOAD_NT"
                 :: "v"(l1), "v"(off1), "s"(xbase) : "memory");
    asm volatile("global_load_async_to_lds_b128 %0, %1, %2 offset:16 th:TH_LOAD_NT"
                 :: "v"(l1), "v"(off1), "s"(xbase) : "memory");
  };

  // Read back one staged tile (4x ds_load_b128), reduce, NT-store b128.
  // max-of-4 written as fmax(fmax(fmax(a,b),c),d) so LLVM folds the first
  // three into v_max3_f32 (2 VALU per output element).
  auto consume = [&](float* o, unsigned stage) {
    const v4f* sp = (const v4f*)(&sbuf[stage][tid][0]);
    const v4f a0 = sp[0];  // row0 cols 0..3
    const v4f a1 = sp[1];  // row0 cols 4..7
    const v4f b0 = sp[2];  // row1 cols 0..3
    const v4f b1 = sp[3];  // row1 cols 4..7
    v4f res;
    res.x = fmaxf(fmaxf(fmaxf(a0.x, a0.y), b0.x), b0.y);
    res.y = fmaxf(fmaxf(fmaxf(a0.z, a0.w), b0.z), b0.w);
    res.z = fmaxf(fmaxf(fmaxf(a1.x, a1.y), b1.x), b1.y);
    res.w = fmaxf(fmaxf(fmaxf(a1.z, a1.w), b1.z), b1.w);
    __builtin_nontemporal_store(res, (v4f*)o);
  };

  issue(inOff, 0);
#pragma unroll
  for (unsigned i = 0; i < kIter; ++i) {
    if (i + 1 < kIter) {
      issue(inOff + kInStepBytes, (i + 1) & 1u);
      // Async loads complete in order: <=4 outstanding => stage i landed.
      asm volatile("s_wait_asynccnt 4" ::: "memory");
    } else {
      asm volatile("s_wait_asynccnt 0" ::: "memory");
    }
    consume(op, i & 1u);
    inOff += kInStepBytes;
    op    += kOutStepFloats;
  }
}

extern "C" void kernel_launch(void* const* d_in, const int* in_sizes, int n_in,
                              void* d_out, int out_size, void* d_ws, size_t ws_size,
                              hipStream_t stream) {
  (void)in_sizes; (void)n_in; (void)out_size; (void)d_ws; (void)ws_size;
  const float* x = (const float*)d_in[0];
  float* out = (float*)d_out;
  maxpool2x2_s2_async_kernel<<<dim3(kBlocks), dim3(kThreads), 0, stream>>>(x, out);
}